// MaxPoolOverPoints_43989055046141
// MI455X (gfx1250) — compile-verified
//
#include <hip/hip_runtime.h>
#include <hip/hip_bf16.h>
#include <cstdint>

// ---------------------------------------------------------------------------
// MaxPoolOverPoints: ui[c,i] = max_k u[c, nb_idx[i,k]] (masked), + pts copy.
//
// MI455X reasoning: u is 12.8 MB -> fully L2-resident (192 MB). Gathered
// payload is N*K*C*4 = 614 MB of L2 traffic; HBM traffic is only ~38 MB
// (~1.6 us at 23.3 TB/s), so this is an L2-gather problem, not a FLOP one.
// Plan: transpose u to (N,64) so each neighbor's channel vector is 256
// contiguous bytes (16x global_load_b128 per lane), keep control flow fully
// uniform by redirecting masked-out neighbors to a -inf dummy row (row N)
// via v_cndmask (dummy row stays hot in L0 for every wave), and use 32-bit
// byte offsets so gathers lower to SGPR-base + VGPR-offset addressing.
// WMMA/TDM do not apply: max-reduce is not A*B+C, and TDM descriptors are
// wave-uniform SGPRs while indices here are per-lane.
// ---------------------------------------------------------------------------

#define CDNA5_C 64
#define CDNA5_K 48

__device__ __forceinline__ float4 f4max(float4 a, float4 b) {
    a.x = fmaxf(a.x, b.x);
    a.y = fmaxf(a.y, b.y);
    a.z = fmaxf(a.z, b.z);
    a.w = fmaxf(a.w, b.w);
    return a;
}

// ---------------------------------------------------------------------------
// Kernel 1: tiled transpose u (C=64, N) -> ut (N+1, 64). 32x32 LDS tiles.
// ---------------------------------------------------------------------------
__global__ __launch_bounds__(256) void transpose_u_kernel(
    const float* __restrict__ u, float* __restrict__ ut, int n)
{
    __shared__ float tile[32][33];
    const int tx = threadIdx.x;   // 0..31
    const int ty = threadIdx.y;   // 0..7
    const int x  = blockIdx.x * 32 + tx;          // point index for loads
    const int c0 = blockIdx.y * 32;               // channel tile base (0 or 32)

    #pragma unroll
    for (int j = 0; j < 32; j += 8) {
        const int c = c0 + ty + j;                // always < 64
        tile[ty + j][tx] = (x < n) ? u[(size_t)c * n + x] : 0.0f;
    }
    __syncthreads();

    const int c  = c0 + tx;                       // channel for stores
    const int y0 = blockIdx.x * 32;
    #pragma unroll
    for (int j = 0; j < 32; j += 8) {
        const int y = y0 + ty + j;                // point index for stores
        if (y < n) ut[(size_t)y * CDNA5_C + c] = tile[tx][ty + j];
    }
}

// Fill dummy row N of ut with -inf (identity under max). One tiny block.
__global__ __launch_bounds__(64) void fill_dummy_row_kernel(
    float* __restrict__ ut, int n)
{
    ut[(size_t)n * CDNA5_C + threadIdx.x] = -INFINITY;
}

// ---------------------------------------------------------------------------
// Kernel 2 (fast path): lane-per-point gather-max from transposed ut.
// Uniform control flow; 32-bit byte offsets (ut < 16 MB) for GVS addressing.
// ---------------------------------------------------------------------------
__global__ __launch_bounds__(256) void maxpool_tr_kernel(
    const float* __restrict__ ut,
    const int*   __restrict__ nb_idx,
    const unsigned char* __restrict__ nb_mask,
    float* __restrict__ out, int n)
{
    const int i = blockIdx.x * blockDim.x + threadIdx.x;
    if (i >= n) return;

    float4 acc[CDNA5_C / 4];
    #pragma unroll
    for (int j = 0; j < CDNA5_C / 4; ++j)
        acc[j] = make_float4(-INFINITY, -INFINITY, -INFINITY, -INFINITY);

    const char* utb   = (const char*)ut;               // byte base (SGPR pair)
    const char* idxb  = (const char*)nb_idx;
    const char* maskb = (const char*)nb_mask;
    const unsigned ibase = (unsigned)i * (CDNA5_K * 4u);   // idx row byte offset
    const unsigned mbase = (unsigned)i * (unsigned)CDNA5_K; // mask row byte offset
    const unsigned dummy_off = (unsigned)n << 8;            // dummy row byte offset

    // software pipeline: pre-load group 0
    int4 nid = *(const int4*)(idxb + ibase);
    unsigned int nm = *(const unsigned int*)(maskb + mbase);

    #pragma unroll 1
    for (int k4 = 0; k4 < CDNA5_K / 4; ++k4) {
        const int4 cur = nid;
        const unsigned int mc = nm;
        if (k4 + 1 < CDNA5_K / 4) {
            nid = *(const int4*)(idxb + ibase + (unsigned)(k4 + 1) * 16u);
            nm  = *(const unsigned int*)(maskb + mbase + (unsigned)(k4 + 1) * 4u);
            // gfx1250: lowers to global_prefetch_b8 — warm next rows toward L0
            __builtin_prefetch(utb + ((unsigned)nid.x << 8), 0, 3);
            __builtin_prefetch(utb + ((unsigned)nid.y << 8), 0, 3);
            __builtin_prefetch(utb + ((unsigned)nid.z << 8), 0, 3);
            __builtin_prefetch(utb + ((unsigned)nid.w << 8), 0, 3);
        }
        // branch-free: invalid neighbors -> dummy -inf row (v_cndmask on offset)
        unsigned off[4];
        off[0] = ((mc >>  0) & 1u) ? ((unsigned)cur.x << 8) : dummy_off;
        off[1] = ((mc >>  8) & 1u) ? ((unsigned)cur.y << 8) : dummy_off;
        off[2] = ((mc >> 16) & 1u) ? ((unsigned)cur.z << 8) : dummy_off;
        off[3] = ((mc >> 24) & 1u) ? ((unsigned)cur.w << 8) : dummy_off;
        #pragma unroll
        for (int e = 0; e < 4; ++e) {
            const float4* __restrict__ row = (const float4*)(utb + off[e]);
            #pragma unroll
            for (int j = 0; j < CDNA5_C / 4; ++j)
                acc[j] = f4max(acc[j], row[j]);
        }
    }

    // out is (C, N): coalesced across lanes for each channel (32-bit offsets)
    float* outf = out;
    const unsigned un = (unsigned)n;
    #pragma unroll
    for (int j = 0; j < CDNA5_C / 4; ++j) {
        outf[(unsigned)(4 * j + 0) * un + (unsigned)i] = acc[j].x;
        outf[(unsigned)(4 * j + 1) * un + (unsigned)i] = acc[j].y;
        outf[(unsigned)(4 * j + 2) * un + (unsigned)i] = acc[j].z;
        outf[(unsigned)(4 * j + 3) * un + (unsigned)i] = acc[j].w;
    }
}

// ---------------------------------------------------------------------------
// Fallback (no workspace): gather directly from u (C,N) with stride-n loads.
// ---------------------------------------------------------------------------
__global__ __launch_bounds__(256) void maxpool_direct_kernel(
    const float* __restrict__ u,
    const int*   __restrict__ nb_idx,
    const unsigned char* __restrict__ nb_mask,
    float* __restrict__ out, int n)
{
    const int i = blockIdx.x * blockDim.x + threadIdx.x;
    if (i >= n) return;

    float acc[CDNA5_C];
    #pragma unroll
    for (int c = 0; c < CDNA5_C; ++c) acc[c] = -INFINITY;

    const int* idxp = nb_idx + (size_t)i * CDNA5_K;
    const unsigned char* mskp = nb_mask + (size_t)i * CDNA5_K;

    #pragma unroll 1
    for (int k = 0; k < CDNA5_K; ++k) {
        if (mskp[k]) {
            const int id = idxp[k];
            #pragma unroll
            for (int c = 0; c < CDNA5_C; ++c)
                acc[c] = fmaxf(acc[c], u[(size_t)c * n + id]);
        }
    }
    #pragma unroll
    for (int c = 0; c < CDNA5_C; ++c)
        out[(size_t)c * n + i] = acc[c];
}

extern "C" void kernel_launch(void* const* d_in, const int* in_sizes, int n_in,
                              void* d_out, int out_size, void* d_ws, size_t ws_size,
                              hipStream_t stream) {
    const float*         u       = (const float*)d_in[0];          // (C, N)
    const float*         pts     = (const float*)d_in[1];          // (N, 2)
    const int*           nb_idx  = (const int*)d_in[2];            // (N, K)
    const unsigned char* nb_mask = (const unsigned char*)d_in[3];  // (N, K) bool
    float* out = (float*)d_out;

    const int n = in_sizes[1] / 2;       // N from pts_x2
    // passthrough output 2: pts_x2, placed after ui (C*N floats)
    hipMemcpyAsync(out + (size_t)CDNA5_C * n, pts,
                   (size_t)in_sizes[1] * sizeof(float),
                   hipMemcpyDeviceToDevice, stream);

    const size_t need = ((size_t)n + 1) * CDNA5_C * sizeof(float);
    const int threads = 256;
    const int blocks  = (n + threads - 1) / threads;

    if (ws_size >= need) {
        float* ut = (float*)d_ws;        // (N+1, 64): transposed u + -inf row
        dim3 tb(32, 8, 1);
        dim3 tg((n + 31) / 32, CDNA5_C / 32, 1);
        transpose_u_kernel<<<tg, tb, 0, stream>>>(u, ut, n);
        fill_dummy_row_kernel<<<1, CDNA5_C, 0, stream>>>(ut, n);
        maxpool_tr_kernel<<<blocks, threads, 0, stream>>>(ut, nb_idx, nb_mask, out, n);
    } else {
        maxpool_direct_kernel<<<blocks, threads, 0, stream>>>(u, nb_idx, nb_mask, out, n);
    }
}